// SHA_39084202393602
// MI455X (gfx1250) — compile-verified
//
#include <hip/hip_runtime.h>
#include <hip/hip_bf16.h>
#include <math.h>

// Flash-decoding style single-query attention for MI455X (gfx1250, wave32).
//   M = 131072 KV rows, DK = DV = 256, fp32 in/out.
// Kernel 1: each block handles `rows` consecutive KV rows:
//   - logits via V_WMMA_F32_16X16X4_F32 (q broadcast across B columns)
//   - local softmax partials (max, sum of exp)
//   - partial weighted-V accumulation (thread t owns column t -> coalesced V reads)
// Kernel 2: merges the per-block partials (log-sum-exp combine) and normalizes.

typedef float v2f __attribute__((ext_vector_type(2)));
typedef float v8f __attribute__((ext_vector_type(8)));

#define DKV 256

__global__ __launch_bounds__(256) void SHA_attn_partial(
    const float* __restrict__ q, const float* __restrict__ K,
    const float* __restrict__ V, float* __restrict__ ws_max,
    float* __restrict__ ws_sum, float* __restrict__ ws_acc, int rows)
{
    __shared__ float q_s[DKV];        // staged query
    __shared__ float logit_s[2048];   // logits -> overwritten with exp-weights
    __shared__ float red_s[256];      // block reduction scratch

    const int tid  = threadIdx.x;
    const int lane = tid & 31;
    const int wave = tid >> 5;
    const long base = (long)blockIdx.x * (long)rows;   // first KV row of this block

    // stage q into LDS (blockDim == DK == 256)
    q_s[tid] = q[tid];
    __syncthreads();

    // ---------------- Phase A: logits = K_chunk . q via WMMA f32 16x16x4 ----
    // A fragment (16x4 f32): lanes 0-15 hold K[row][k0+0..1], lanes 16-31 hold
    // K[row][k0+2..3] (row = lane&15). B fragment broadcasts q so every D
    // column is the same dot product; accumulate over 64 k-steps.
    const int ngroups = rows >> 4;            // 16 rows per group
    const int sub = (lane < 16) ? 0 : 2;      // per-lane k sub-offset
    const int r   = lane & 15;                // row within group
    for (int g = wave; g < ngroups; g += 8) {
        const float* kp = K + (base + (long)g * 16 + r) * DKV + sub;
        v8f c = {};
        #pragma unroll 8
        for (int k0 = 0; k0 < DKV; k0 += 4) {
            v2f a = *(const v2f*)(kp + k0);           // global_load_b64 (K)
            v2f b = *(const v2f*)(&q_s[k0 + sub]);    // ds_load_b64 (q broadcast)
            c = __builtin_amdgcn_wmma_f32_16x16x4_f32(
                    false, a, false, b, (short)0, c, false, false);
        }
        // lanes 0 and 16 hold rows g*16+0..7 / +8..15 in c[0..7]
        if (r == 0) {
            float* dst = &logit_s[g * 16 + (lane >> 4) * 8];
            #pragma unroll
            for (int i = 0; i < 8; ++i) dst[i] = c[i];
        }
    }
    __syncthreads();

    // ---------------- Phase B: local softmax partials -----------------------
    float lmax = -INFINITY;
    for (int m = tid; m < rows; m += 256) lmax = fmaxf(lmax, logit_s[m]);
    red_s[tid] = lmax;
    __syncthreads();
    for (int s = 128; s > 0; s >>= 1) {
        if (tid < s) red_s[tid] = fmaxf(red_s[tid], red_s[tid + s]);
        __syncthreads();
    }
    const float bmax = red_s[0];
    __syncthreads();

    float lsum = 0.f;
    for (int m = tid; m < rows; m += 256) {
        float e = expf(logit_s[m] - bmax);
        logit_s[m] = e;                    // keep weights for phase C
        lsum += e;
    }
    red_s[tid] = lsum;
    __syncthreads();
    for (int s = 128; s > 0; s >>= 1) {
        if (tid < s) red_s[tid] += red_s[tid + s];
        __syncthreads();
    }
    const float bsum = red_s[0];

    // ---------------- Phase C: partial acc[t] = sum_m w[m] * V[m][t] --------
    // thread t owns output column t -> fully coalesced 1KB V row reads.
    float acc = 0.f;
    const float* vp = V + base * DKV + tid;
    for (int m = 0; m < rows; m += 4) {
        float4 w4 = *(const float4*)&logit_s[m];   // LDS broadcast read
        __builtin_prefetch(vp + (long)(m + 16) * DKV, 0, 1);
        acc += w4.x * vp[(long)(m + 0) * DKV];
        acc += w4.y * vp[(long)(m + 1) * DKV];
        acc += w4.z * vp[(long)(m + 2) * DKV];
        acc += w4.w * vp[(long)(m + 3) * DKV];
    }
    ws_acc[(long)blockIdx.x * DKV + tid] = acc;
    if (tid == 0) { ws_max[blockIdx.x] = bmax; ws_sum[blockIdx.x] = bsum; }
}

__global__ __launch_bounds__(256) void SHA_attn_reduce(
    const float* __restrict__ ws_max, const float* __restrict__ ws_sum,
    const float* __restrict__ ws_acc, float* __restrict__ out, int nblk)
{
    __shared__ float red_s[256];
    __shared__ float scale_s[512];
    const int tid = threadIdx.x;

    float lmax = -INFINITY;
    for (int b = tid; b < nblk; b += 256) lmax = fmaxf(lmax, ws_max[b]);
    red_s[tid] = lmax;
    __syncthreads();
    for (int s = 128; s > 0; s >>= 1) {
        if (tid < s) red_s[tid] = fmaxf(red_s[tid], red_s[tid + s]);
        __syncthreads();
    }
    const float gmax = red_s[0];
    __syncthreads();

    float lsum = 0.f;
    for (int b = tid; b < nblk; b += 256) {
        float sc = expf(ws_max[b] - gmax);
        scale_s[b] = sc;
        lsum += sc * ws_sum[b];
    }
    red_s[tid] = lsum;
    __syncthreads();
    for (int s = 128; s > 0; s >>= 1) {
        if (tid < s) red_s[tid] += red_s[tid + s];
        __syncthreads();
    }
    const float total = red_s[0];

    float y = 0.f;
    for (int b = 0; b < nblk; ++b)
        y += scale_s[b] * ws_acc[(long)b * DKV + tid];   // coalesced
    out[tid] = y / total;
}

extern "C" void kernel_launch(void* const* d_in, const int* in_sizes, int n_in,
                              void* d_out, int out_size, void* d_ws, size_t ws_size,
                              hipStream_t stream)
{
    const float* q = (const float*)d_in[0];   // (256,)
    const float* K = (const float*)d_in[1];   // (131072, 256)
    const float* V = (const float*)d_in[2];   // (131072, 256)
    float* out = (float*)d_out;               // (256,)

    const int M = in_sizes[1] / DKV;          // 131072

    // Pick the split so partials fit in d_ws (per block: 2 + 256 floats).
    int nblk = 512;
    while (nblk > 64 && (size_t)nblk * (DKV + 2) * sizeof(float) > ws_size)
        nblk >>= 1;
    const int rows = M / nblk;                // 256 .. 2048 (<= logit_s capacity)

    float* ws_max = (float*)d_ws;
    float* ws_sum = ws_max + nblk;
    float* ws_acc = ws_sum + nblk;

    SHA_attn_partial<<<nblk, 256, 0, stream>>>(q, K, V, ws_max, ws_sum, ws_acc, rows);
    SHA_attn_reduce<<<1, 256, 0, stream>>>(ws_max, ws_sum, ws_acc, out, nblk);
}